// Model_91036126806540
// MI455X (gfx1250) — compile-verified
//
#include <hip/hip_runtime.h>
#include <hip/hip_bf16.h>

// GQA decode attention, split-KV flash-decode, fp32 WMMA (V_WMMA_F32_16X16X4_F32).
// Pass 1: B*HKV*NSPLIT blocks, each owns a 1024-row KV chunk (load-balanced,
//         high occupancy); 8 waves per block split the chunk; the 16 query
//         rows (g=4 heads x SQ=4) are the WMMA M dimension. Partial (O,m,l)
//         go to workspace.
// Pass 2: tiny merge kernel combines the NSPLIT partials per (b,hkv).
// Fallback: if ws_size is too small, launch the fused NSPLIT=1 variant.

#define B_      32
#define SQ_     4
#define SKV_    4096
#define HQ_     32
#define HKV_    8
#define G_      4
#define D_      128
#define NW      8           // waves per block
#define NSPLIT_ 4           // KV splits per (b, hkv) in pass 1
#define NEGV    (-1.0e30f)

typedef float v2f __attribute__((ext_vector_type(2)));
typedef float v8f __attribute__((ext_vector_type(8)));

static __device__ __forceinline__ v8f wmma_f32_16x16x4(v2f a, v2f b, v8f c) {
  // (neg_a, A, neg_b, B, c_mod, C, reuse_a, reuse_b)
  return __builtin_amdgcn_wmma_f32_16x16x4_f32(false, a, false, b, (short)0, c,
                                               false, false);
}

template <int NSPLIT>
__global__ __launch_bounds__(256)
void gqa_fd_kernel(const float* __restrict__ q,
                   const float* __restrict__ kc,
                   const float* __restrict__ vc,
                   const int* __restrict__ seqlens,
                   float* __restrict__ out,     // used when NSPLIT==1
                   float* __restrict__ wsO,     // [B*HKV][NSPLIT][16][128]
                   float* __restrict__ wsM,     // [B*HKV][NSPLIT][16]
                   float* __restrict__ wsL) {   // [B*HKV][NSPLIT][16]
  __shared__ float O_sh[16 * D_];        // merged partial-O accumulator (8 KB)
  __shared__ float P_sh[NW][16 * 16];    // per-wave P layout-shuffle buffer
  __shared__ float m_sh[NW][16];
  __shared__ float l_sh[NW][16];
  __shared__ float Mtot[16];
  __shared__ float Ltot[16];

  const int tid   = threadIdx.x;
  const int lane  = tid & 31;
  const int wave  = tid >> 5;
  const int row16 = lane & 15;     // N-index (C/D layout) / M-index (A layout)
  const int half  = lane >> 4;     // 0: lanes 0-15, 1: lanes 16-31
  const int koff  = half * 2;      // K sub-offset for A/B fragments

  const int idx   = blockIdx.x;
  const int bh    = idx / NSPLIT;
  const int split = idx - bh * NSPLIT;
  const int b  = bh >> 3;          // / HKV_
  const int hk = bh & 7;           // % HKV_

  const int seqlen  = seqlens[b];
  const int chunk   = SKV_ / NSPLIT;
  const int kvstart = split * chunk;
  const int kvend0  = kvstart + chunk;
  const int kvend   = seqlen < kvend0 ? seqlen : kvend0;
  const int ntiles  = kvend > kvstart ? ((kvend - kvstart + 15) >> 4) : 0;

  // zero shared accumulator (d_out/d_ws are poisoned; we own init)
  for (int i = tid; i < 16 * D_; i += 256) O_sh[i] = 0.0f;

  // ---- Q A-fragment, resident in VGPRs, pre-scaled by 1/sqrt(D) ----
  const float scale = 0.08838834764831845f;   // 1/sqrt(128)
  const int mA = row16;                       // query row held by this lane
  const int gi = mA >> 2;                     // head within GQA group
  const int s  = mA & 3;                      // query position
  const int h  = hk * G_ + gi;
  const float* qptr = q + (((size_t)b * SQ_ + s) * HQ_ + h) * D_;

  v2f aq[32];
#pragma unroll
  for (int kk = 0; kk < 32; ++kk) {
    v2f t = *(const v2f*)(qptr + 4 * kk + koff);
    aq[kk].x = t.x * scale;
    aq[kk].y = t.y * scale;
  }

  const size_t kvstride = (size_t)HKV_ * D_;  // 1024 floats between kv rows
  const float* kbase = kc + ((size_t)b * SKV_ * HKV_ + hk) * D_;
  const float* vbase = vc + ((size_t)b * SKV_ * HKV_ + hk) * D_;

  const v8f zero8 = {};
  v8f  o[8];                                  // O tile: 8 d-chunks of 16
  float mr[8], lr[8];                         // online-softmax state per row
#pragma unroll
  for (int j = 0; j < 8; ++j) { o[j] = zero8; mr[j] = NEGV; lr[j] = 0.0f; }

  float* pbuf = &P_sh[wave][0];

  // ---- flash-decode main loop: this wave owns tiles wave, wave+NW, ... ----
  for (int t = wave; t < ntiles; t += NW) {
    const int kv0 = kvstart + (t << 4);

    // prefetch this wave's next tile (~128 rows ahead)
    const int kvn = kv0 + (NW << 4);
    if (kvn < kvend) {
      __builtin_prefetch(kbase + (size_t)(kvn + row16) * kvstride, 0, 0);
      __builtin_prefetch(vbase + (size_t)(kvn + row16) * kvstride, 0, 0);
    }

    // S = Q * K^T : 32 chained 16x16x4 fp32 WMMAs over D=128
    const float* kp = kbase + (size_t)(kv0 + row16) * kvstride;
    v8f c = zero8;
#pragma unroll
    for (int kk = 0; kk < 32; ++kk) {
      v2f bk = *(const v2f*)(kp + 4 * kk + koff);
      c = wmma_f32_16x16x4(aq[kk], bk, c);
    }

    // mask + online softmax; column index == kv0+row16 for all 8 C VGPRs
    const bool valid = (kv0 + row16) < kvend;
#pragma unroll
    for (int j = 0; j < 8; ++j) {
      float x = valid ? c[j] : NEGV;
      float xm = x;                           // row-max across 16 lanes
      xm = fmaxf(xm, __shfl_xor(xm, 1, 32));
      xm = fmaxf(xm, __shfl_xor(xm, 2, 32));
      xm = fmaxf(xm, __shfl_xor(xm, 4, 32));
      xm = fmaxf(xm, __shfl_xor(xm, 8, 32));
      const float mnew  = fmaxf(mr[j], xm);
      const float alpha = __expf(mr[j] - mnew);
      const float p     = __expf(x - mnew);   // masked lanes -> exp(-1e30)=0
      float ps = p;                           // row-sum across 16 lanes
      ps += __shfl_xor(ps, 1, 32);
      ps += __shfl_xor(ps, 2, 32);
      ps += __shfl_xor(ps, 4, 32);
      ps += __shfl_xor(ps, 8, 32);
      lr[j] = lr[j] * alpha + ps;
      mr[j] = mnew;
#pragma unroll
      for (int nc = 0; nc < 8; ++nc) o[nc][j] *= alpha;
      // stash P in row-major [m][n] for A-layout reload (same wave, DS waits)
      pbuf[(j + 8 * half) * 16 + row16] = p;
    }

    // A-fragments of P for the PV matmul (K=16 -> 4 k-steps)
    v2f pa[4];
#pragma unroll
    for (int ks = 0; ks < 4; ++ks)
      pa[ks] = *(const v2f*)(pbuf + mA * 16 + 4 * ks + koff);

    // O += P * V : 8 d-chunks x 4 k-steps of 16x16x4 fp32 WMMA
    const float* vp = vbase + (size_t)kv0 * kvstride;
#pragma unroll
    for (int nc = 0; nc < 8; ++nc) {
#pragma unroll
      for (int ks = 0; ks < 4; ++ks) {
        v2f bv;
        bv.x = vp[(size_t)(4 * ks + koff)     * kvstride + nc * 16 + row16];
        bv.y = vp[(size_t)(4 * ks + koff + 1) * kvstride + nc * 16 + row16];
        o[nc] = wmma_f32_16x16x4(pa[ks], bv, o[nc]);
      }
    }
  }

  // ---- merge the NW waves' partial (m, l, O) within the block ----
  if (row16 == 0) {
#pragma unroll
    for (int j = 0; j < 8; ++j) {
      m_sh[wave][j + 8 * half] = mr[j];
      l_sh[wave][j + 8 * half] = lr[j];
    }
  }
  __syncthreads();

#pragma unroll
  for (int j = 0; j < 8; ++j) {
    const int row = j + 8 * half;
    float M = NEGV;
#pragma unroll
    for (int w = 0; w < NW; ++w) M = fmaxf(M, m_sh[w][row]);
    float L = 0.0f;
#pragma unroll
    for (int w = 0; w < NW; ++w) L += l_sh[w][row] * __expf(m_sh[w][row] - M);
    const float sw = __expf(mr[j] - M);
#pragma unroll
    for (int nc = 0; nc < 8; ++nc)
      atomicAdd(&O_sh[row * D_ + nc * 16 + row16], o[nc][j] * sw);  // ds_add_f32
    if (wave == 0 && row16 == 0) { Mtot[row] = M; Ltot[row] = L; }
  }
  __syncthreads();

  if constexpr (NSPLIT == 1) {
    // fused path: out[b][s][h][d] = O / L, coalesced
    for (int i = tid; i < 16 * D_; i += 256) {
      const int row = i >> 7;
      const int d   = i & 127;
      const int gi2 = row >> 2;
      const int s2  = row & 3;
      const int h2  = hk * G_ + gi2;
      out[(((size_t)b * SQ_ + s2) * HQ_ + h2) * D_ + d] = O_sh[i] / Ltot[row];
    }
  } else {
    // split path: write partials (always, even for empty chunks: O=0, m=-1e30, l=0)
    const size_t base = (size_t)bh * NSPLIT + split;
    float* po = wsO + base * (16 * D_);
    for (int i = tid; i < 16 * D_; i += 256) po[i] = O_sh[i];
    if (tid < 16) {
      wsM[base * 16 + tid] = Mtot[tid];
      wsL[base * 16 + tid] = Ltot[tid];
    }
  }
}

__global__ __launch_bounds__(256)
void gqa_merge_kernel(const float* __restrict__ wsO,
                      const float* __restrict__ wsM,
                      const float* __restrict__ wsL,
                      float* __restrict__ out) {
  const int bh = blockIdx.x;
  const int b  = bh >> 3;
  const int hk = bh & 7;
  const int tid = threadIdx.x;

  for (int i = tid; i < 16 * D_; i += 256) {
    const int row = i >> 7;
    const int d   = i & 127;
    float m[NSPLIT_], l[NSPLIT_];
    float M = NEGV;
#pragma unroll
    for (int sp = 0; sp < NSPLIT_; ++sp) {
      m[sp] = wsM[((size_t)bh * NSPLIT_ + sp) * 16 + row];
      l[sp] = wsL[((size_t)bh * NSPLIT_ + sp) * 16 + row];
      M = fmaxf(M, m[sp]);
    }
    float num = 0.0f, den = 0.0f;
#pragma unroll
    for (int sp = 0; sp < NSPLIT_; ++sp) {
      const float f = __expf(m[sp] - M);     // empty chunk: exp(-1e30-M)=0
      num += wsO[((size_t)bh * NSPLIT_ + sp) * (16 * D_) + i] * f;
      den += l[sp] * f;
    }
    const int gi2 = row >> 2;
    const int s2  = row & 3;
    const int h2  = hk * G_ + gi2;
    out[(((size_t)b * SQ_ + s2) * HQ_ + h2) * D_ + d] = num / den;
  }
}

extern "C" void kernel_launch(void* const* d_in, const int* in_sizes, int n_in,
                              void* d_out, int out_size, void* d_ws, size_t ws_size,
                              hipStream_t stream) {
  const float* q  = (const float*)d_in[0];
  const float* kc = (const float*)d_in[1];
  const float* vc = (const float*)d_in[2];
  const int*   sl = (const int*)d_in[3];
  float* out = (float*)d_out;
  (void)in_sizes; (void)n_in; (void)out_size;

  const size_t nBH   = (size_t)B_ * HKV_;                 // 256
  const size_t nO    = nBH * NSPLIT_ * 16 * D_;           // partial O floats
  const size_t nML   = nBH * NSPLIT_ * 16;                // per-row stats floats
  const size_t need  = (nO + 2 * nML) * sizeof(float);    // ~8.5 MB

  if (ws_size >= need) {
    float* wsO = (float*)d_ws;
    float* wsM = wsO + nO;
    float* wsL = wsM + nML;
    hipLaunchKernelGGL((gqa_fd_kernel<NSPLIT_>), dim3(nBH * NSPLIT_), dim3(256),
                       0, stream, q, kc, vc, sl, out, wsO, wsM, wsL);
    hipLaunchKernelGGL(gqa_merge_kernel, dim3(nBH), dim3(256),
                       0, stream, wsO, wsM, wsL, out);
  } else {
    // deterministic fallback: fused single-pass variant
    hipLaunchKernelGGL((gqa_fd_kernel<1>), dim3(nBH), dim3(256),
                       0, stream, q, kc, vc, sl, out,
                       (float*)nullptr, (float*)nullptr, (float*)nullptr);
  }
}